// PartialDerivator_91293824844467
// MI455X (gfx1250) — compile-verified
//
#include <hip/hip_runtime.h>
#include <math.h>

// Eikonal partial-derivative kernel for MI455X (gfx1250, wave32).
// Gather/bandwidth-bound scalar workload: no WMMA-shaped contraction exists
// (all math is 2x2 quadratic forms + per-vertex masked argmin). CDNA5-specific
// optimization: the per-workgroup adjacency slab (256 rows x 96 B = 24 KB,
// perfectly contiguous) is DMA'd into LDS with the Tensor Data Mover
// (tensor_load_to_lds + s_wait_tensorcnt), then read as ds_load_b128 -- one
// linear DMA stream instead of 6 strided b96 gathers per lane. Irregular
// gathers (solution, vertices, tensor field) stay as global loads: they are
// L2-resident (all inputs ~82 MB < 192 MB L2). global_prefetch_b8 warms the
// next adjacency slab.

#define EIK_EPS 1e-8f
#define ADJ 6
#define BLOCK 256

#if __has_builtin(__builtin_amdgcn_tensor_load_to_lds) && \
    __has_builtin(__builtin_amdgcn_s_wait_tensorcnt)
#define USE_TDM 1
#else
#define USE_TDM 0
#endif

typedef unsigned int v4u __attribute__((ext_vector_type(4)));
typedef int v8i __attribute__((ext_vector_type(8)));
typedef int v4i __attribute__((ext_vector_type(4)));

__device__ __forceinline__ float quadf(float ex, float ey,
                                       float m00, float m01, float m10, float m11) {
  // e . (M e)  for general (not assumed symmetric) M, matching einsum('i,ij,j')
  return ex * (m00 * ex + m01 * ey) + ey * (m10 * ex + m11 * ey);
}

__device__ __forceinline__ float interior_value(float lr,
    float uj, float uk, float e1x, float e1y, float e2x, float e2y,
    float m00, float m01, float m10, float m11) {
  float lam = fminf(fmaxf(lr, 0.0f), 1.0f);
  float px = lam * e1x + (1.0f - lam) * e2x;
  float py = lam * e1y + (1.0f - lam) * e2y;
  float T = fmaxf(quadf(px, py, m00, m01, m10, m11), EIK_EPS);
  return lam * uj + (1.0f - lam) * uk + sqrtf(T);
}

// Computes the two quadratic-root candidates' raw lambdas + validity.
__device__ __forceinline__ void interior_lams(
    float uj, float uk, float e1x, float e1y, float e2x, float e2y,
    float m00, float m01, float m10, float m11,
    float& lamp, float& lamm, bool& okp, bool& okm) {
  float du = uj - uk;
  float dx = e1x - e2x, dy = e1y - e2y;
  float a = quadf(dx, dy, m00, m01, m10, m11);
  float b = e2x * (m00 * dx + m01 * dy) + e2y * (m10 * dx + m11 * dy);
  float c = quadf(e2x, e2y, m00, m01, m10, m11);
  float du2 = du * du;
  float A_ = a - du2;
  float Aq = a * A_;
  float Bq = 2.0f * b * A_;
  float Cq = b * b - du2 * c;
  float disc = Bq * Bq - 4.0f * Aq * Cq;
  bool pos = disc > 0.0f;
  float sq = pos ? sqrtf(disc) : 0.0f;
  float Aqs = (fabsf(Aq) < EIK_EPS) ? EIK_EPS : Aq;
  float inv2A = 1.0f / (2.0f * Aqs);
  lamp = (-Bq + sq) * inv2A;
  lamm = (-Bq - sq) * inv2A;
  okp = pos && (lamp > 0.0f) && (lamp < 1.0f);
  okm = pos && (lamm > 0.0f) && (lamm < 1.0f);
}

// Hand-differentiated _interior_value (matches JAX where-gated autodiff).
// Accumulates w * grad into gs[2] (d/du_j, d/du_k) and gm[4] (row-major 2x2).
__device__ __forceinline__ void interior_grad(
    float uj, float uk, float e1x, float e1y, float e2x, float e2y,
    float m00, float m01, float m10, float m11, float sign,
    float w, float* gs, float* gm) {
  float du = uj - uk;
  float dx = e1x - e2x, dy = e1y - e2y;
  float a = quadf(dx, dy, m00, m01, m10, m11);
  float b = e2x * (m00 * dx + m01 * dy) + e2y * (m10 * dx + m11 * dy);
  float c = quadf(e2x, e2y, m00, m01, m10, m11);
  float du2 = du * du;
  float A_ = a - du2;
  float Aq = a * A_;
  float Bq = 2.0f * b * A_;
  float Cq = b * b - du2 * c;
  float disc = Bq * Bq - 4.0f * Aq * Cq;
  bool pos = disc > 0.0f;
  float sq = pos ? sqrtf(disc) : 0.0f;
  bool aqsmall = fabsf(Aq) < EIK_EPS;
  float Aqs = aqsmall ? EIK_EPS : Aq;
  float inv2A = 1.0f / (2.0f * Aqs);
  float lr = (-Bq + sign * sq) * inv2A;
  float dclip = (lr > 0.0f && lr < 1.0f) ? 1.0f : 0.0f;   // clip gradient gate
  float lam = fminf(fmaxf(lr, 0.0f), 1.0f);

  // sq = where(disc>0, sqrt(disc), 0): gradient flows only when disc>0
  float invsq = pos ? (1.0f / sq) : 0.0f;
  float dsq_dAq = pos ? (-2.0f * Cq * invsq) : 0.0f;
  float dsq_dBq = pos ? (Bq * invsq) : 0.0f;
  float dsq_dCq = pos ? (-2.0f * Aq * invsq) : 0.0f;

  float dlr_dBq = (-1.0f + sign * dsq_dBq) * inv2A;
  float dlr_dCq = (sign * dsq_dCq) * inv2A;
  // Aq_safe gate: no gradient to Aq when |Aq|<EPS
  float dlr_dAq = (sign * dsq_dAq) * inv2A - (aqsmall ? 0.0f : lr / Aqs);

  float dlr_da = dlr_dAq * (2.0f * a - du2) + dlr_dBq * (2.0f * b);
  float dlr_db = dlr_dBq * (2.0f * A_) + dlr_dCq * (2.0f * b);
  float dlr_dc = dlr_dCq * (-du2);
  float dlr_ddu = dlr_dAq * (-2.0f * a * du) + dlr_dBq * (-4.0f * b * du)
                + dlr_dCq * (-2.0f * du * c);

  float dlam_da = dclip * dlr_da, dlam_db = dclip * dlr_db;
  float dlam_dc = dclip * dlr_dc, dlam_ddu = dclip * dlr_ddu;

  float px = lam * e1x + (1.0f - lam) * e2x;
  float py = lam * e1y + (1.0f - lam) * e2y;
  float pMp = quadf(px, py, m00, m01, m10, m11);
  float T = fmaxf(pMp, EIK_EPS);
  float sT = sqrtf(T);
  float indT = (pMp > EIK_EPS) ? 1.0f : 0.0f;             // max(.,EPS) gate
  float hinv = indT * 0.5f / sT;                          // d value / d pMp

  // d(pMp)/dlam = p^T (M + M^T) d
  float msum = m01 + m10;
  float pMsd = px * (2.0f * m00 * dx + msum * dy) + py * (msum * dx + 2.0f * m11 * dy);
  float Glam = du + hinv * pMsd;                          // d value / d lam

  gs[0] += w * (lam + Glam * dlam_ddu);
  gs[1] += w * ((1.0f - lam) - Glam * dlam_ddu);

  float cdd   = Glam * dlam_da;   // coefficient of d (x) d
  float ce2d  = Glam * dlam_db;   // coefficient of e2 (x) d
  float ce2e2 = Glam * dlam_dc;   // coefficient of e2 (x) e2
  gm[0] += w * (cdd * dx * dx + ce2d * e2x * dx + ce2e2 * e2x * e2x + hinv * px * px);
  gm[1] += w * (cdd * dx * dy + ce2d * e2x * dy + ce2e2 * e2x * e2y + hinv * px * py);
  gm[2] += w * (cdd * dy * dx + ce2d * e2y * dx + ce2e2 * e2y * e2x + hinv * py * px);
  gm[3] += w * (cdd * dy * dy + ce2d * e2y * dy + ce2e2 * e2y * e2y + hinv * py * py);
}

__global__ __launch_bounds__(BLOCK)
void eikonal_partials_kernel(const float* __restrict__ u,
                             const float* __restrict__ tens,   // (S,2,2) row-major
                             const float* __restrict__ verts,  // (N,2)
                             const int*   __restrict__ adj,    // (N,6,4)
                             float* __restrict__ gsol,         // (N,6,2)
                             float* __restrict__ gpar,         // (N,6,4)
                             int N) {
  const int block_start = blockIdx.x * BLOCK;
  const int rows = min(BLOCK, N - block_start);     // >= 1 (grid sized to N)
  const int n = block_start + threadIdx.x;

#if USE_TDM
  // ---- TDM: DMA the block's contiguous 24KB adjacency slab into LDS ----
  __shared__ __align__(16) int lds_adj[BLOCK * 4 * ADJ];
  if (threadIdx.x == 0) {
    unsigned long long gaddr =
        (unsigned long long)(uintptr_t)adj + (unsigned long long)block_start * (4ull * ADJ * 4ull);
    unsigned lds_base = (unsigned)(uintptr_t)&lds_adj[0];   // low 32 bits = LDS offset
    unsigned d0 = (unsigned)rows * (4u * ADJ);              // dwords in slab (<= 6144)

    v4u g0;
    g0[0] = 1u;                                             // count=1, user descriptor
    g0[1] = lds_base;                                       // lds_addr
    g0[2] = (unsigned)gaddr;                                // global_addr[31:0]
    g0[3] = (unsigned)((gaddr >> 32) & 0x1FFFFFFull)        // global_addr[56:32]
          | 0x80000000u;                                    // type=2 ("image")

    v8i g1;
    g1[0] = 0x00020000;                  // workgroup_mask=0, data_size=2 (4B), no pad/iter
    g1[1] = (int)((d0 & 0xFFFFu) << 16); // atomic_barrier_addr=0, tensor_dim0[15:0]
    g1[2] = (int)((d0 >> 16) & 0xFFFFu)  // tensor_dim0[31:16]
          | (1 << 16);                   // tensor_dim1 = 1
    g1[3] = (int)((d0 & 0xFFFFu) << 16); // tensor_dim1 hi=0, tile_dim0 = d0 (fits 16b)
    g1[4] = 1;                           // tile_dim1 = 1, tile_dim2 = 0
    g1[5] = (int)d0;                     // tensor_dim0_stride[31:0]
    g1[6] = 0;                           // stride hi, dim1_stride lo
    g1[7] = 0;

    v4i g2 = {0, 0, 0, 0};               // <=2D tensor: groups 2/3 unused
    v4i g3 = {0, 0, 0, 0};
    v8i g4 = {0, 0, 0, 0, 0, 0, 0, 0};   // extra descriptor group (clang-23 6-arg form)

    __builtin_amdgcn_tensor_load_to_lds(g0, g1, g2, g3, g4, 0);
    __builtin_amdgcn_s_wait_tensorcnt(0);                   // TENSORcnt -> 0
  }
  __syncthreads();
#endif

  if (n < N) {
    // gfx1250 prefetch path (global_prefetch_b8): warm the next adjacency slab
    if (n + 4096 < N) __builtin_prefetch(&adj[(size_t)(n + 4096) * 4 * ADJ], 0, 1);

#if USE_TDM
    const int4* arow = (const int4*)&lds_adj[(size_t)threadIdx.x * 4 * ADJ];
#else
    const int4* arow = (const int4*)(adj + (size_t)n * 4 * ADJ);
#endif
    int jj[ADJ], kk[ADJ], ss[ADJ];
#pragma unroll
    for (int a = 0; a < ADJ; ++a) {
      int4 q = arow[a];                 // ds_load_b128 (TDM path) / b128 global
      jj[a] = q.y; kk[a] = q.z; ss[a] = q.w;
    }

    float vix = verts[2 * (size_t)n];
    float viy = verts[2 * (size_t)n + 1];

    // ---- pass 1: 24 masked candidate values, per-vertex min ----
    float vals[ADJ][4];
    float minv = __builtin_inff();
#pragma unroll
    for (int a = 0; a < ADJ; ++a) {
      bool vad = ss[a] >= 0;
      int sA = vad ? ss[a] : 0;
      float uj = u[jj[a]], uk = u[kk[a]];
      float vjx = verts[2 * (size_t)jj[a]], vjy = verts[2 * (size_t)jj[a] + 1];
      float vkx = verts[2 * (size_t)kk[a]], vky = verts[2 * (size_t)kk[a] + 1];
      float e1x = vix - vjx, e1y = viy - vjy;
      float e2x = vix - vkx, e2y = viy - vky;
      float4 M = ((const float4*)tens)[sA];  // b128 load of 2x2 tensor
      float m00 = M.x, m01 = M.y, m10 = M.z, m11 = M.w;

      float s0 = sqrtf(fmaxf(quadf(e2x, e2y, m00, m01, m10, m11), EIK_EPS));
      float s1 = sqrtf(fmaxf(quadf(e1x, e1y, m00, m01, m10, m11), EIK_EPS));
      float lamp, lamm; bool okp, okm;
      interior_lams(uj, uk, e1x, e1y, e2x, e2y, m00, m01, m10, m11,
                    lamp, lamm, okp, okm);

      const float INF = __builtin_inff();
      float v0 = vad ? (uk + s0) : INF;
      float v1 = vad ? (uj + s1) : INF;
      float vp = (vad && okp)
          ? interior_value(lamp, uj, uk, e1x, e1y, e2x, e2y, m00, m01, m10, m11) : INF;
      float vm = (vad && okm)
          ? interior_value(lamm, uj, uk, e1x, e1y, e2x, e2y, m00, m01, m10, m11) : INF;
      vals[a][0] = v0; vals[a][1] = v1; vals[a][2] = vp; vals[a][3] = vm;
      minv = fminf(minv, fminf(fminf(v0, v1), fminf(vp, vm)));
    }

    // tie count: softmin degenerates to indicator (exp(-k*0)=1 on ties)
    int cnt = 0;
#pragma unroll
    for (int a = 0; a < ADJ; ++a)
#pragma unroll
      for (int c = 0; c < 4; ++c) cnt += (vals[a][c] == minv) ? 1 : 0;
    float w = 1.0f / fmaxf((float)cnt, EIK_EPS);

    // ---- pass 2: tie-gated gradient recompute + coalesced stores ----
#pragma unroll
    for (int a = 0; a < ADJ; ++a) {
      float gs[2] = {0.0f, 0.0f};
      float gm[4] = {0.0f, 0.0f, 0.0f, 0.0f};
      bool t0 = vals[a][0] == minv, t1 = vals[a][1] == minv;
      bool t2 = vals[a][2] == minv, t3 = vals[a][3] == minv;
      if ((t0 | t1 | t2 | t3) && ss[a] >= 0) {
        int sA = ss[a];
        float uj = u[jj[a]], uk = u[kk[a]];
        float vjx = verts[2 * (size_t)jj[a]], vjy = verts[2 * (size_t)jj[a] + 1];
        float vkx = verts[2 * (size_t)kk[a]], vky = verts[2 * (size_t)kk[a] + 1];
        float e1x = vix - vjx, e1y = viy - vjy;
        float e2x = vix - vkx, e2y = viy - vky;
        float4 M = ((const float4*)tens)[sA];
        float m00 = M.x, m01 = M.y, m10 = M.z, m11 = M.w;

        if (t0) {  // val0 = u_k + sqrt(max(e2.M.e2, eps)); gu=(0,1); gM=e2(x)e2/(2 s0)
          float s0 = sqrtf(fmaxf(quadf(e2x, e2y, m00, m01, m10, m11), EIK_EPS));
          float h = w * 0.5f / s0;
          gs[1] += w;
          gm[0] += h * e2x * e2x; gm[1] += h * e2x * e2y;
          gm[2] += h * e2y * e2x; gm[3] += h * e2y * e2y;
        }
        if (t1) {  // val1 = u_j + sqrt(max(e1.M.e1, eps)); gu=(1,0); gM=e1(x)e1/(2 s1)
          float s1 = sqrtf(fmaxf(quadf(e1x, e1y, m00, m01, m10, m11), EIK_EPS));
          float h = w * 0.5f / s1;
          gs[0] += w;
          gm[0] += h * e1x * e1x; gm[1] += h * e1x * e1y;
          gm[2] += h * e1y * e1x; gm[3] += h * e1y * e1y;
        }
        if (t2) interior_grad(uj, uk, e1x, e1y, e2x, e2y, m00, m01, m10, m11,
                              +1.0f, w, gs, gm);
        if (t3) interior_grad(uj, uk, e1x, e1y, e2x, e2y, m00, m01, m10, m11,
                              -1.0f, w, gs, gm);
      }
      *(float2*)(gsol + ((size_t)n * ADJ + a) * 2) = make_float2(gs[0], gs[1]);
      *(float4*)(gpar + ((size_t)n * ADJ + a) * 4) = make_float4(gm[0], gm[1], gm[2], gm[3]);
    }
  }
}

extern "C" void kernel_launch(void* const* d_in, const int* in_sizes, int n_in,
                              void* d_out, int out_size, void* d_ws, size_t ws_size,
                              hipStream_t stream) {
  const float* u     = (const float*)d_in[0];   // solution_vector (N)
  const float* tens  = (const float*)d_in[1];   // tensor_field (S,2,2)
  const float* verts = (const float*)d_in[2];   // vertices (N,2)
  const int*   adj   = (const int*)d_in[3];     // adjacent_vertex_inds (N,6,4)
  int N = in_sizes[0];
  float* gsol = (float*)d_out;                        // (N,6,2)
  float* gpar = (float*)d_out + (size_t)N * ADJ * 2;  // (N,6,2,2)
  int blocks = (N + BLOCK - 1) / BLOCK;
  hipLaunchKernelGGL(eikonal_partials_kernel, dim3(blocks), dim3(BLOCK), 0, stream,
                     u, tens, verts, adj, gsol, gpar, N);
}